// DSSMamba_47296179863957
// MI455X (gfx1250) — compile-verified
//
#include <hip/hip_runtime.h>
#include <hip/hip_bf16.h>
#include <math.h>

// Bidirectional Mamba for MI455X (gfx1250, wave32).
// GEMMs: V_WMMA_F32_16X16X4_F32 (fp32 matrix path, matches reference numerics).
// Selective scan: chunked parallel scan (64 chunks x 64 steps) -> 512 blocks,
// critical path 3*64 steps instead of 4096. B/C broadcasts are uniform loads
// (scalar-load candidates) -> no barriers in the scan inner loop.

typedef __attribute__((ext_vector_type(2))) float v2f;
typedef __attribute__((ext_vector_type(8))) float v8f;

#define BATCH 4
#define SEQLEN 4096
#define DMODEL 256
#define DINNER 512
#define DTRANK 16
#define DSTATE 16
#define MROWS (BATCH * SEQLEN)                 // 16384
#define XZ_COLS (4 * DINNER)                   // 2048  [x_f, z_f, x_b, z_b]
#define XDBL_COLS (2 * (DTRANK + 2 * DSTATE))  // 96
#define DELTA_COLS (2 * DINNER)                // 1024
#define Y_COLS (2 * DINNER)                    // 1024
#define NCHUNK 64
#define CLEN (SEQLEN / NCHUNK)                 // 64
#define STATE_STRIDE (DINNER * DSTATE)         // 8192 floats per (bd, chunk)

__device__ __forceinline__ float silu_f(float x) {
    return x / (1.0f + __expf(-x));
}

__device__ __forceinline__ v8f wmma4(v2f a, v2f b, v8f c) {
    return __builtin_amdgcn_wmma_f32_16x16x4_f32(false, a, false, b, (short)0, c,
                                                 false, false);
}

// ---------------------------------------------------------------------------
// K1: xz[m, n] = sum_k h[m,k] * Wi[n,k]   (n<1024 -> Wi_f, else Wi_b)
// grid (16, 1024), block 256 (8 waves). LDS-staged 16x256 A tile per block.
// ---------------------------------------------------------------------------
__global__ void k_gemm_xz(const float* __restrict__ h,
                          const float* __restrict__ Wi_f,
                          const float* __restrict__ Wi_b,
                          float* __restrict__ xz) {
    __shared__ float lA[16 * 256];
    const int m0 = blockIdx.y * 16;
    for (int i = 0; i < 16; ++i)
        lA[i * 256 + threadIdx.x] = h[(size_t)(m0 + i) * DMODEL + threadIdx.x];
    __syncthreads();

    const int wave = threadIdx.x >> 5;
    const int lane = threadIdx.x & 31;
    const int g = lane >> 4, ln = lane & 15;
    const int ng = (blockIdx.x * 8 + wave) * 16;        // 0..2047
    const float* W = (ng < 2 * DINNER) ? Wi_f : Wi_b;
    const int nb = (ng < 2 * DINNER) ? ng : (ng - 2 * DINNER);

    v8f acc = {0.f, 0.f, 0.f, 0.f, 0.f, 0.f, 0.f, 0.f};
    const float* wrow = &W[(size_t)(nb + ln) * DMODEL];
    for (int k0 = 0; k0 < DMODEL; k0 += 4) {
        v2f a = *(const v2f*)&lA[ln * 256 + k0 + 2 * g];
        v2f b = *(const v2f*)&wrow[k0 + 2 * g];
        acc = wmma4(a, b, acc);
    }
#pragma unroll
    for (int v = 0; v < 8; ++v)
        xz[(size_t)(m0 + v + 8 * g) * XZ_COLS + ng + ln] = acc[v];
}

// ---------------------------------------------------------------------------
// K2: x_dbl[m, dir*48 + n] = sum_k silu(x_dir[m,k]) * Wx_dir[n,k]  (n<48)
// grid 1024, block 256. 6 compute waves (2 dirs x 3 N-tiles); silu applied
// while staging 16x128 K-chunks of x into LDS.
// ---------------------------------------------------------------------------
__global__ void k_gemm_xdbl(const float* __restrict__ xz,
                            const float* __restrict__ Wx_f,
                            const float* __restrict__ Wx_b,
                            float* __restrict__ xdbl) {
    __shared__ float lU[2 * 16 * 128];
    const int m0 = blockIdx.x * 16;
    const int wave = threadIdx.x >> 5;
    const int lane = threadIdx.x & 31;
    const int g = lane >> 4, ln = lane & 15;
    const int dir = wave / 3;             // waves 6,7 idle for compute
    const int n0 = (wave % 3) * 16;
    const float* Wx = (dir == 1) ? Wx_b : Wx_f;

    v8f acc = {0.f, 0.f, 0.f, 0.f, 0.f, 0.f, 0.f, 0.f};
    for (int kc = 0; kc < DINNER; kc += 128) {
        for (int i = 0; i < 16; ++i) {
            int e = i * 256 + threadIdx.x;        // 0..4095
            int d_ = e >> 11;                     // dir
            int r = (e >> 7) & 15;                // row in tile
            int c = e & 127;                      // col in chunk
            const float* src = &xz[(size_t)(m0 + r) * XZ_COLS + d_ * (2 * DINNER) + kc + c];
            if (kc + 128 < DINNER)
                __builtin_prefetch(src + 128, 0, 1);   // global_prefetch_b8
            lU[e] = silu_f(*src);
        }
        __syncthreads();
        if (wave < 6) {
            const float* wrow = &Wx[(size_t)(n0 + ln) * DINNER + kc];
            for (int k0 = 0; k0 < 128; k0 += 4) {
                v2f a = *(const v2f*)&lU[dir * 2048 + ln * 128 + k0 + 2 * g];
                v2f b = *(const v2f*)&wrow[k0 + 2 * g];
                acc = wmma4(a, b, acc);
            }
        }
        __syncthreads();
    }
    if (wave < 6) {
#pragma unroll
        for (int v = 0; v < 8; ++v)
            xdbl[(size_t)(m0 + v + 8 * g) * XDBL_COLS + dir * 48 + n0 + ln] = acc[v];
    }
}

// ---------------------------------------------------------------------------
// K3: delta[m, dir*512 + d] = softplus( sum_r xdbl[m, dir*48 + r]*Wdt[d,r] + bdt[d] )
// grid 8192, block 256. One wave per 16x16 tile, K=16 -> 4 WMMA ops.
// ---------------------------------------------------------------------------
__global__ void k_delta(const float* __restrict__ xdbl,
                        const float* __restrict__ Wdt_f,
                        const float* __restrict__ Wdt_b,
                        const float* __restrict__ bdt_f,
                        const float* __restrict__ bdt_b,
                        float* __restrict__ delta) {
    const int wave = threadIdx.x >> 5;
    const int lane = threadIdx.x & 31;
    const int g = lane >> 4, ln = lane & 15;
    const int tile = blockIdx.x * 8 + wave;       // 0..65535
    const int mt = tile >> 6;
    const int rest = tile & 63;
    const int dir = rest >> 5;
    const int n0 = (rest & 31) * 16;
    const int m0 = mt * 16;
    const float* Wdt = dir ? Wdt_b : Wdt_f;
    const float* bdt = dir ? bdt_b : bdt_f;

    v8f acc = {0.f, 0.f, 0.f, 0.f, 0.f, 0.f, 0.f, 0.f};
    for (int k0 = 0; k0 < DTRANK; k0 += 4) {
        v2f a = *(const v2f*)&xdbl[(size_t)(m0 + ln) * XDBL_COLS + dir * 48 + k0 + 2 * g];
        v2f b = *(const v2f*)&Wdt[(size_t)(n0 + ln) * DTRANK + k0 + 2 * g];
        acc = wmma4(a, b, acc);
    }
    const float bias = bdt[n0 + ln];
#pragma unroll
    for (int v = 0; v < 8; ++v) {
        float val = acc[v] + bias;
        float sp = (val > 20.0f) ? val : log1pf(__expf(val));
        delta[(size_t)(m0 + v + 8 * g) * DELTA_COLS + dir * DINNER + n0 + ln] = sp;
    }
}

// ---------------------------------------------------------------------------
// K4a: chunk-local scan summaries. grid 512 = (batch, dir, chunk), block 512.
// Computes P_c[n] = prod exp(dt*A) and S_c[n] = chunk-local state from 0.
// B row is block-uniform -> scalar-load broadcast, no LDS, no barriers.
// ---------------------------------------------------------------------------
__global__ void k_scan_chunk(const float* __restrict__ xz,
                             const float* __restrict__ xdbl,
                             const float* __restrict__ delta,
                             const float* __restrict__ A_log_f,
                             const float* __restrict__ A_log_b,
                             float* __restrict__ Pbuf,
                             float* __restrict__ Sbuf) {
    const int c = blockIdx.x % NCHUNK;
    const int bd = blockIdx.x / NCHUNK;            // 0..7 = b*2 + dir
    const int b = bd >> 1, dir = bd & 1;
    const int d = threadIdx.x;                     // 0..511
    const float* A_log = dir ? A_log_b : A_log_f;

    float A[DSTATE];
#pragma unroll
    for (int n = 0; n < DSTATE; ++n)
        A[n] = -__expf(A_log[d * DSTATE + n]);
    float s[DSTATE], pr[DSTATE];
#pragma unroll
    for (int n = 0; n < DSTATE; ++n) { s[n] = 0.f; pr[n] = 1.f; }

    for (int j = 0; j < CLEN; ++j) {
        const int t = c * CLEN + j;
        const int p = dir ? (SEQLEN - 1 - t) : t;
        const size_t row = (size_t)b * SEQLEN + p;
        const float* bc = &xdbl[row * XDBL_COLS + dir * 48 + DTRANK]; // uniform

        const float dt = delta[row * DELTA_COLS + dir * DINNER + d];
        const float xv = xz[row * XZ_COLS + dir * (2 * DINNER) + d];
        if (j + 1 < CLEN) {
            const int pn = dir ? (p - 1) : (p + 1);
            const size_t rn = (size_t)b * SEQLEN + pn;
            __builtin_prefetch(&delta[rn * DELTA_COLS + dir * DINNER + d], 0, 1);
            __builtin_prefetch(&xz[rn * XZ_COLS + dir * (2 * DINNER) + d], 0, 1);
        }
        const float du = dt * silu_f(xv);
#pragma unroll
        for (int n = 0; n < DSTATE; ++n) {
            float a = __expf(dt * A[n]);
            s[n] = a * s[n] + du * bc[n];
            pr[n] *= a;
        }
    }
    const size_t base = ((size_t)bd * NCHUNK + c) * STATE_STRIDE + (size_t)d * DSTATE;
#pragma unroll
    for (int n = 0; n < DSTATE; ++n) {
        Pbuf[base + n] = pr[n];
        Sbuf[base + n] = s[n];
    }
}

// ---------------------------------------------------------------------------
// K4b: combine chunk summaries sequentially over NCHUNK (tiny critical path).
// 65536 independent (b,dir,d,n) threads: Sin_c = run; run = P_c*run + S_c.
// ---------------------------------------------------------------------------
__global__ void k_scan_carry(const float* __restrict__ Pbuf,
                             const float* __restrict__ Sbuf,
                             float* __restrict__ Sin) {
    const int idx = blockIdx.x * blockDim.x + threadIdx.x;  // 0..65535
    const int bd = idx / STATE_STRIDE;
    const int rem = idx % STATE_STRIDE;
    float run = 0.f;
    for (int c = 0; c < NCHUNK; ++c) {
        const size_t e = ((size_t)bd * NCHUNK + c) * STATE_STRIDE + rem;
        Sin[e] = run;
        run = Pbuf[e] * run + Sbuf[e];
    }
}

// ---------------------------------------------------------------------------
// K4c: re-run each chunk from its true initial state; emit gated outputs.
// grid 512 = (batch, dir, chunk), block 512. Uniform B/C loads, no barriers.
// ---------------------------------------------------------------------------
__global__ void k_scan_apply(const float* __restrict__ xz,
                             const float* __restrict__ xdbl,
                             const float* __restrict__ delta,
                             const float* __restrict__ A_log_f,
                             const float* __restrict__ A_log_b,
                             const float* __restrict__ D_f,
                             const float* __restrict__ D_b,
                             const float* __restrict__ Sin,
                             float* __restrict__ yall) {
    const int c = blockIdx.x % NCHUNK;
    const int bd = blockIdx.x / NCHUNK;
    const int b = bd >> 1, dir = bd & 1;
    const int d = threadIdx.x;
    const float* A_log = dir ? A_log_b : A_log_f;
    const float Dp = (dir ? D_b : D_f)[d];

    float A[DSTATE];
#pragma unroll
    for (int n = 0; n < DSTATE; ++n)
        A[n] = -__expf(A_log[d * DSTATE + n]);
    float s[DSTATE];
    const size_t base = ((size_t)bd * NCHUNK + c) * STATE_STRIDE + (size_t)d * DSTATE;
#pragma unroll
    for (int n = 0; n < DSTATE; ++n) s[n] = Sin[base + n];

    for (int j = 0; j < CLEN; ++j) {
        const int t = c * CLEN + j;
        const int p = dir ? (SEQLEN - 1 - t) : t;
        const size_t row = (size_t)b * SEQLEN + p;
        const float* bc = &xdbl[row * XDBL_COLS + dir * 48 + DTRANK]; // uniform

        const float dt = delta[row * DELTA_COLS + dir * DINNER + d];
        const float xv = xz[row * XZ_COLS + dir * (2 * DINNER) + d];
        const float zv = xz[row * XZ_COLS + dir * (2 * DINNER) + DINNER + d];
        if (j + 1 < CLEN) {
            const int pn = dir ? (p - 1) : (p + 1);
            const size_t rn = (size_t)b * SEQLEN + pn;
            __builtin_prefetch(&delta[rn * DELTA_COLS + dir * DINNER + d], 0, 1);
            __builtin_prefetch(&xz[rn * XZ_COLS + dir * (2 * DINNER) + d], 0, 1);
        }
        const float u = silu_f(xv);
        const float du = dt * u;
        float y = 0.f;
#pragma unroll
        for (int n = 0; n < DSTATE; ++n) {
            float a = __expf(dt * A[n]);
            s[n] = a * s[n] + du * bc[n];
            y += s[n] * bc[DSTATE + n];
        }
        y += u * Dp;
        y *= silu_f(zv);
        yall[row * Y_COLS + dir * DINNER + d] = y;
    }
}

// ---------------------------------------------------------------------------
// K5: out[m, n] = sum_k yall[m,k] * Wo[n,k].  grid (2, 1024), block 256.
// LDS-staged 16x256 K-chunks of y shared by 8 waves.
// ---------------------------------------------------------------------------
__global__ void k_gemm_out(const float* __restrict__ yall,
                           const float* __restrict__ Wo,
                           float* __restrict__ out) {
    __shared__ float lY[16 * 256];
    const int m0 = blockIdx.y * 16;
    const int wave = threadIdx.x >> 5;
    const int lane = threadIdx.x & 31;
    const int g = lane >> 4, ln = lane & 15;
    const int n0 = (blockIdx.x * 8 + wave) * 16;   // 0..255

    v8f acc = {0.f, 0.f, 0.f, 0.f, 0.f, 0.f, 0.f, 0.f};
    for (int kc = 0; kc < Y_COLS; kc += 256) {
        for (int i = 0; i < 16; ++i) {
            const float* src = &yall[(size_t)(m0 + i) * Y_COLS + kc + threadIdx.x];
            if (kc + 256 < Y_COLS)
                __builtin_prefetch(src + 256, 0, 1);
            lY[i * 256 + threadIdx.x] = *src;
        }
        __syncthreads();
        const float* wrow = &Wo[(size_t)(n0 + ln) * Y_COLS + kc];
        for (int k0 = 0; k0 < 256; k0 += 4) {
            v2f a = *(const v2f*)&lY[ln * 256 + k0 + 2 * g];
            v2f b = *(const v2f*)&wrow[k0 + 2 * g];
            acc = wmma4(a, b, acc);
        }
        __syncthreads();
    }
#pragma unroll
    for (int v = 0; v < 8; ++v)
        out[(size_t)(m0 + v + 8 * g) * DMODEL + n0 + ln] = acc[v];
}

// ---------------------------------------------------------------------------
extern "C" void kernel_launch(void* const* d_in, const int* in_sizes, int n_in,
                              void* d_out, int out_size, void* d_ws, size_t ws_size,
                              hipStream_t stream) {
    const float* h      = (const float*)d_in[0];
    const float* Wi_f   = (const float*)d_in[1];
    const float* Wi_b   = (const float*)d_in[2];
    const float* Wx_f   = (const float*)d_in[3];
    const float* Wx_b   = (const float*)d_in[4];
    const float* Wdt_f  = (const float*)d_in[5];
    const float* Wdt_b  = (const float*)d_in[6];
    const float* bdt_f  = (const float*)d_in[7];
    const float* bdt_b  = (const float*)d_in[8];
    const float* A_logf = (const float*)d_in[9];
    const float* A_logb = (const float*)d_in[10];
    const float* D_f    = (const float*)d_in[11];
    const float* D_b    = (const float*)d_in[12];
    const float* Wo     = (const float*)d_in[13];
    float* out = (float*)d_out;

    // Workspace partition (floats). Total ~325 MB.
    float* ws    = (float*)d_ws;
    float* xz    = ws;                                        // 16384*2048
    float* xdbl  = xz    + (size_t)MROWS * XZ_COLS;           // 16384*96
    float* delta = xdbl  + (size_t)MROWS * XDBL_COLS;         // 16384*1024
    float* yall  = delta + (size_t)MROWS * DELTA_COLS;        // 16384*1024
    float* Pbuf  = yall  + (size_t)MROWS * Y_COLS;            // 8*64*8192
    float* Sbuf  = Pbuf  + (size_t)8 * NCHUNK * STATE_STRIDE; // 8*64*8192
    float* Sin   = Sbuf  + (size_t)8 * NCHUNK * STATE_STRIDE; // 8*64*8192

    k_gemm_xz<<<dim3(16, MROWS / 16), 256, 0, stream>>>(h, Wi_f, Wi_b, xz);
    k_gemm_xdbl<<<MROWS / 16, 256, 0, stream>>>(xz, Wx_f, Wx_b, xdbl);
    k_delta<<<(MROWS / 16) * 2 * (DINNER / 16) / 8, 256, 0, stream>>>(
        xdbl, Wdt_f, Wdt_b, bdt_f, bdt_b, delta);
    k_scan_chunk<<<8 * NCHUNK, DINNER, 0, stream>>>(xz, xdbl, delta,
                                                    A_logf, A_logb, Pbuf, Sbuf);
    k_scan_carry<<<(8 * STATE_STRIDE) / 512, 512, 0, stream>>>(Pbuf, Sbuf, Sin);
    k_scan_apply<<<8 * NCHUNK, DINNER, 0, stream>>>(xz, xdbl, delta, A_logf, A_logb,
                                                    D_f, D_b, Sin, yall);
    k_gemm_out<<<dim3(2, MROWS / 16), 256, 0, stream>>>(yall, Wo, out);
}